// MultiHeadHyperedgeAttention_20220706030434
// MI455X (gfx1250) — compile-verified
//
#include <hip/hip_runtime.h>
#include <math.h>

typedef __attribute__((ext_vector_type(2))) float v2f;
typedef __attribute__((ext_vector_type(8))) float v8f;

#define NUM_NODES      100000
#define NUM_EDGES      640000
#define NUM_HE         50000
#define IN_DIM         128
#define NUM_HEADS      8
#define HEAD_DIM       8
#define N_OUT          64                 // NUM_HEADS * HEAD_DIM
#define NTILES         ((NUM_HE + 15) / 16)   // 3125 (exact: 50000 = 3125*16)
#define WAVES_PER_BLK  4

// ---------------------------------------------------------------------------
// Kernel 1: zero workspace (counts[NUM_HE] followed by sums[NUM_HE*IN_DIM])
// ---------------------------------------------------------------------------
__global__ void zero_ws_kernel(float* __restrict__ ws, int n) {
    int i = blockIdx.x * blockDim.x + threadIdx.x;
    int stride = gridDim.x * blockDim.x;
    for (; i < n; i += stride) ws[i] = 0.0f;
}

// ---------------------------------------------------------------------------
// Kernel 2: scatter-add gathered node features into hyperedge sums.
// One wave per edge; lane c handles float4 chunk c of the 128-float row.
// Gather of x is fully coalesced (32 lanes read 512 consecutive bytes).
// ---------------------------------------------------------------------------
__global__ void scatter_kernel(const float4* __restrict__ x4,
                               const int* __restrict__ node_idx,
                               const int* __restrict__ he_idx,
                               float* __restrict__ sums,
                               float* __restrict__ counts) {
    long long t = (long long)blockIdx.x * blockDim.x + threadIdx.x;
    const long long total = (long long)NUM_EDGES * 32;
    if (t >= total) return;
    int e = (int)(t >> 5);
    int c = (int)(t & 31);
    int node = node_idx[e];
    int he   = he_idx[e];
    float4 v = x4[(size_t)node * 32 + c];
    float* dst = sums + (size_t)he * IN_DIM + c * 4;
    atomicAdd(dst + 0, v.x);
    atomicAdd(dst + 1, v.y);
    atomicAdd(dst + 2, v.z);
    atomicAdd(dst + 3, v.w);
    if (c == 0) atomicAdd(counts + he, 1.0f);
}

// ---------------------------------------------------------------------------
// Kernel 3: feats = sums/max(count,1); H = relu(feats@W1 + b1);
//           alpha = per-head dot(H, W2) + b2; out = mean(sigmoid(clip)) * .9 + .1
// Per wave: one 16-row tile, N=64 split into 4 WMMA column tiles, K=128 in
// steps of 4 via V_WMMA_F32_16X16X4_F32 (fp32 matrix path).
//
// LDS B layout is K-PAIRED: pair p = d/2 holds (B[2p][n], B[2p+1][n]) in
// adjacent dwords, so every v2f B fragment is one aligned ds_load_b64
// (kb is always even: kb = k0 + 2*half, k0 % 4 == 0). A fragments are a
// single global_load_b64 + v_pk_mul_f32 by 1/count.
// ---------------------------------------------------------------------------
__global__ __launch_bounds__(WAVES_PER_BLK * 32)
void he_attn_wmma_kernel(const float* __restrict__ sums,
                         const float* __restrict__ counts,
                         const float* __restrict__ W1,   // [H, D, K] = [8,128,8]
                         const float* __restrict__ b1,   // [8,8]
                         const float* __restrict__ W2,   // [8,8]
                         const float* __restrict__ b2,   // [8]
                         float* __restrict__ out) {      // [NUM_HE]
    // ldsBp[p*64 + n] = (B[2p][n], B[2p+1][n]),  p = 0..63, n = 0..63   (32 KB)
    __shared__ v2f   ldsBp[(IN_DIM / 2) * N_OUT];
    __shared__ float stage[WAVES_PER_BLK * 16 * N_OUT];                  // 16 KB

    const int tid = threadIdx.x;

    // Pre-transform W1[h,d,k] -> K-paired B layout in LDS. n = h*8+k.
    {
        float* ldsBf = (float*)ldsBp;
        for (int i = tid; i < IN_DIM * N_OUT; i += blockDim.x) {
            int d = i >> 6;
            int n = i & 63;
            // element (d, n) lives at pair (d>>1), slot (d&1)
            ldsBf[((d >> 1) * 64 + n) * 2 + (d & 1)] =
                W1[((n >> 3) << 10) + (d << 3) + (n & 7)];
        }
    }
    __syncthreads();

    const int wave = tid >> 5;
    const int lane = tid & 31;
    int tile = blockIdx.x * WAVES_PER_BLK + wave;
    const bool valid = (tile < NTILES);
    if (!valid) tile = NTILES - 1;          // clamp: keep barriers/EXEC uniform

    const int rowbase = tile * 16;
    const int mrow = rowbase + (lane & 15); // A layout: lanes hold M = lane&15
    const int half = lane >> 4;             // A layout: K pair selected by lane>>4
    const int ncol = lane & 15;             // B/C layout: N = lane&15

    float cnt  = counts[mrow];
    float invc = 1.0f / fmaxf(cnt, 1.0f);
    const v2f* __restrict__ arow2 = (const v2f*)(sums + (size_t)mrow * IN_DIM);

    v8f acc0 = {}, acc1 = {}, acc2 = {}, acc3 = {};

    for (int k0 = 0; k0 < IN_DIM; k0 += 4) {
        const int kb = k0 + half * 2;       // even
        const int p  = kb >> 1;             // K-pair index
        // A fragment: (feats[mrow][kb], feats[mrow][kb+1]) = one b64 load
        v2f A = arow2[p];
        A *= invc;
        // B fragments: one ds_load_b64 each
        v2f B0 = ldsBp[p * 64 +  0 + ncol];
        v2f B1 = ldsBp[p * 64 + 16 + ncol];
        v2f B2 = ldsBp[p * 64 + 32 + ncol];
        v2f B3 = ldsBp[p * 64 + 48 + ncol];

        acc0 = __builtin_amdgcn_wmma_f32_16x16x4_f32(false, A, false, B0, (short)0, acc0, false, false);
        acc1 = __builtin_amdgcn_wmma_f32_16x16x4_f32(false, A, false, B1, (short)0, acc1, false, false);
        acc2 = __builtin_amdgcn_wmma_f32_16x16x4_f32(false, A, false, B2, (short)0, acc2, false, false);
        acc3 = __builtin_amdgcn_wmma_f32_16x16x4_f32(false, A, false, B3, (short)0, acc3, false, false);
    }

    // Stage D tiles to LDS. C/D layout: VGPR r -> row r (lanes 0-15) / r+8 (16-31).
    float* st = stage + wave * (16 * N_OUT);
#pragma unroll
    for (int r = 0; r < 8; r++) {
        const int row_local = r + half * 8;
        st[row_local * 64 +  0 + ncol] = acc0[r];
        st[row_local * 64 + 16 + ncol] = acc1[r];
        st[row_local * 64 + 32 + ncol] = acc2[r];
        st[row_local * 64 + 48 + ncol] = acc3[r];
    }
    __syncthreads();

    // Epilogue: lanes 0..15 each own one output row.
    if (lane < 16) {
        const float* hr = st + lane * 64;
        float avg = 0.0f;
#pragma unroll
        for (int hh = 0; hh < NUM_HEADS; hh++) {
            float alpha = b2[hh];
#pragma unroll
            for (int k = 0; k < HEAD_DIM; k++) {
                const int n = hh * 8 + k;
                float hv = fmaxf(hr[n] + b1[n], 0.0f);   // bias + ReLU
                alpha += hv * W2[n];
            }
            alpha = fminf(fmaxf(alpha, -5.0f), 5.0f);
            avg += 1.0f / (1.0f + expf(-alpha));
        }
        if (valid) out[rowbase + lane] = (avg * 0.125f) * 0.9f + 0.1f;
    }
}

// ---------------------------------------------------------------------------
extern "C" void kernel_launch(void* const* d_in, const int* in_sizes, int n_in,
                              void* d_out, int out_size, void* d_ws, size_t ws_size,
                              hipStream_t stream) {
    const float* x      = (const float*)d_in[0];
    const int*   nidx   = (const int*)  d_in[1];
    const int*   heidx  = (const int*)  d_in[2];
    const float* W1     = (const float*)d_in[3];
    const float* b1     = (const float*)d_in[4];
    const float* W2     = (const float*)d_in[5];
    const float* b2     = (const float*)d_in[6];
    float* out = (float*)d_out;

    float* counts = (float*)d_ws;            // NUM_HE floats
    float* sums   = counts + NUM_HE;         // NUM_HE * IN_DIM floats (16B aligned)

    // 1) zero counts+sums (contiguous)
    const int zn = NUM_HE * (IN_DIM + 1);
    zero_ws_kernel<<<1024, 256, 0, stream>>>(counts, zn);

    // 2) scatter-mean numerators/denominators
    const long long tasks = (long long)NUM_EDGES * 32;
    const int sblocks = (int)((tasks + 255) / 256);
    scatter_kernel<<<sblocks, 256, 0, stream>>>((const float4*)x, nidx, heidx,
                                                sums, counts);

    // 3) WMMA GEMM + attention epilogue
    const int ablocks = (NTILES + WAVES_PER_BLK - 1) / WAVES_PER_BLK;  // 782
    he_attn_wmma_kernel<<<ablocks, WAVES_PER_BLK * 32, 0, stream>>>(
        sums, counts, W1, b1, W2, b2, out);
}